// RoughScorer_5076651344576
// MI455X (gfx1250) — compile-verified
//
#include <hip/hip_runtime.h>
#include <hip/hip_bf16.h>
#include <stdint.h>

// ---------------------------------------------------------------------------
// RoughScorer top-k for MI455X (gfx1250).
//   scores[i][j] = M[i][j] + M[j][i]  (j < i, else -inf);  top-50 per row.
// 32 rows per block, 64-wide column tiles. Row tile A (32x64) and transposed
// tile B (64x32) are DMA'd by the Tensor Data Mover (tensor_load_to_lds),
// double-buffered with s_wait_tensorcnt. D# LDS padding (+1 dword per row)
// gives stride-65 / stride-33 LDS rows -> bank-conflict-free A[r][c], B[c][r].
// Selection: per-row candidate buffer (u32 mono-score + u16 index) with a
// running 50th-score threshold; compaction rows picked via one ballot per
// tile (uniform across waves); block-cooperative bitonic sorts; final fill.
// ---------------------------------------------------------------------------

#define NN   8192
#define KK   50
#define ROWS 32           // rows per block
#define CT   64           // columns per tile
#define CC   128          // candidate capacity per row

// LDS layout (bytes). TDM pad gives A rows 65 dwords, B rows 33 dwords.
#define TILEA_BYTES (32 * 65 * 4)        // 8320
#define TILEB_BYTES (64 * 33 * 4)        // 8448
#define OFF_A0   0
#define OFF_A1   (OFF_A0 + TILEA_BYTES)  // 8320
#define OFF_B0   (OFF_A1 + TILEA_BYTES)  // 16640
#define OFF_B1   (OFF_B0 + TILEB_BYTES)  // 25088
#define OFF_MONO (OFF_B1 + TILEB_BYTES)  // 33536
#define OFF_IDX  (OFF_MONO + ROWS * CC * 4)   // 49920
#define OFF_CNT  (OFF_IDX + ROWS * CC * 2)    // 58112
#define OFF_THR  (OFF_CNT + 128)              // 58240
#define SMEM_BYTES (OFF_THR + 128)            // 58368 (< 64KB)

typedef unsigned int v4u __attribute__((ext_vector_type(4)));
typedef int          v8i __attribute__((ext_vector_type(8)));
typedef int          v4i __attribute__((ext_vector_type(4)));

// ---- monotone float<->uint mapping (order-preserving) ---------------------
__device__ __forceinline__ unsigned mono_from_float(float f) {
  unsigned u = __float_as_uint(f);
  return (u & 0x80000000u) ? ~u : (u | 0x80000000u);
}
__device__ __forceinline__ float float_from_mono(unsigned m) {
  unsigned u = (m & 0x80000000u) ? (m & 0x7FFFFFFFu) : ~m;
  return __uint_as_float(u);
}

// ---- TDM: DMA a 2D f32 tile (row stride NN elems) into LDS ---------------
// D# group0: count=1, lds_addr, global_addr, type=2.
// D# group1: data_size=4B, pad_enable, pad_interval enc (2^(enc+1) dwords),
//            pad_amount enc 0 -> 1 dword; tensor NNxNN, tile d0 x d1.
// Groups 2/3 zero (2D tensor); 6-arg builtin form for this toolchain.
__device__ __forceinline__ void tdm_load_tile(unsigned lds_byte_off,
                                              const float* gaddr,
                                              unsigned tile_d0, unsigned tile_d1,
                                              unsigned pad_enc) {
  unsigned long long ga = (unsigned long long)(uintptr_t)gaddr;
  v4u g0;
  g0[0] = 0x1u;                                   // count=1 (valid), user mode
  g0[1] = lds_byte_off;                           // lds_addr (bytes)
  g0[2] = (unsigned)(ga & 0xFFFFFFFFull);         // global_addr[31:0]
  g0[3] = (unsigned)((ga >> 32) & 0x01FFFFFFull)  // global_addr[56:32]
        | (2u << 30);                             // type=2 ("image")
  v8i g1;
  g1[0] = (int)((2u << 16)         // data_size = 4 bytes
              | (1u << 20)         // pad_enable
              | (pad_enc << 22)    // pad_interval
              | (0u << 25));       // pad_amount: enc 0 -> 1 dword
  g1[1] = (int)((NN & 0xFFFFu) << 16);                            // d0 lo16
  g1[2] = (int)(((NN >> 16) & 0xFFFFu) | ((NN & 0xFFFFu) << 16)); // d0 hi/d1 lo
  g1[3] = (int)(((NN >> 16) & 0xFFFFu) | (tile_d0 << 16));        // d1 hi/tile_d0
  g1[4] = (int)tile_d1;                                           // tile_d1
  g1[5] = (int)NN;                                                // dim0_stride
  g1[6] = 0;
  g1[7] = 0;
  v4i z4; z4[0] = 0; z4[1] = 0; z4[2] = 0; z4[3] = 0;             // groups 2/3
  v8i z8; z8[0] = 0; z8[1] = 0; z8[2] = 0; z8[3] = 0;
          z8[4] = 0; z8[5] = 0; z8[6] = 0; z8[7] = 0;
  __builtin_amdgcn_tensor_load_to_lds(g0, g1, z4, z4, z8, 0);
}

// ---- block-cooperative bitonic sort of one row (desc score, asc index) ----
__device__ __forceinline__ void sort_row_desc(unsigned* mono, unsigned short* idx,
                                              unsigned* cnt, float* thr,
                                              int rr, int tid, bool updThr) {
  unsigned* bm = mono + rr * CC;
  unsigned short* bi = idx + rr * CC;
  unsigned n0 = cnt[rr];
  for (int i = tid; i < CC; i += 256)
    if ((unsigned)i >= n0) { bm[i] = 0u; bi[i] = 0xFFFFu; }  // pads sort last
  __syncthreads();
  for (int k = 2; k <= CC; k <<= 1) {
    for (int j = k >> 1; j > 0; j >>= 1) {
      int i = tid;
      if (i < CC) {
        int l = i ^ j;
        if (l > i) {
          unsigned am = bm[i], lm = bm[l];
          unsigned short ai = bi[i], li = bi[l];
          bool ls = (am < lm) || (am == lm && ai > li);   // i ranks after l
          bool gt = (lm < am) || (lm == am && li > ai);   // i ranks before l
          bool up = ((i & k) == 0);
          if (up ? ls : gt) { bm[i] = lm; bm[l] = am; bi[i] = li; bi[l] = ai; }
        }
      }
      __syncthreads();
    }
  }
  if (tid == 0) {
    unsigned nc = n0 < (unsigned)KK ? n0 : (unsigned)KK;
    cnt[rr] = nc;
    if (updThr && n0 >= (unsigned)KK)
      thr[rr] = float_from_mono(bm[KK - 1]);
  }
  __syncthreads();
}

__global__ void rough_topk_kernel(const float* __restrict__ M,
                                  float* __restrict__ outS,
                                  int* __restrict__ outI) {
  extern __shared__ char smem[];
  unsigned*       mono = (unsigned*)(smem + OFF_MONO);
  unsigned short* idx  = (unsigned short*)(smem + OFF_IDX);
  unsigned*       cnt  = (unsigned*)(smem + OFF_CNT);
  float*          thr  = (float*)(smem + OFF_THR);

  const int tid = (int)threadIdx.x;
  const int blk = (int)blockIdx.x;
  const int i0  = blk * ROWS;
  const int tmax = (i0 + ROWS - 2) / CT + 1;   // tiles with any j < i entries
  const bool issuer = (tid < 32);              // wave 0 drives the TDM

  if (tid < ROWS) { cnt[tid] = 0u; thr[tid] = -__builtin_inff(); }

  if (issuer) {                                // prologue: tile 0 -> buffer 0
    tdm_load_tile(OFF_A0, M + (size_t)i0 * NN, CT, ROWS, 5u);  // 32 rows x 64
    tdm_load_tile(OFF_B0, M + i0,              ROWS, CT, 4u);  // 64 rows x 32
  }
  __syncthreads();

  int buf = 0;
  for (int t = 0; t < tmax; ++t) {
    if (issuer) {
      if (t + 1 < tmax) {                      // prefetch next tile
        const int jt2 = (t + 1) * CT;
        tdm_load_tile(buf ? OFF_A0 : OFF_A1,
                      M + (size_t)i0 * NN + jt2, CT, ROWS, 5u);
        tdm_load_tile(buf ? OFF_B0 : OFF_B1,
                      M + (size_t)jt2 * NN + i0, ROWS, CT, 4u);
        __builtin_amdgcn_s_wait_tensorcnt(2);  // current tile's 2 loads done
      } else {
        __builtin_amdgcn_s_wait_tensorcnt(0);
      }
    }
    __syncthreads();                           // tile visible to all 8 waves

    const float* Abuf = (const float*)(smem + (buf ? OFF_A1 : OFF_A0));
    const float* Bbuf = (const float*)(smem + (buf ? OFF_B1 : OFF_B0));
    const int r  = tid & 31;                   // lane-distinct -> no conflicts
    const int cg = tid >> 5;                   // 0..7 column group
    const int gi = i0 + r;
    const int jt = t * CT;
    const float thv = thr[r];
    const float* Ar = Abuf + r * 65;
#pragma unroll
    for (int s = 0; s < 8; ++s) {
      const int c  = cg * 8 + s;
      const int gj = jt + c;
      if (gi > gj) {
        const float sc = Ar[c] + Bbuf[c * 33 + r];
        if (sc > thv) {
          unsigned pos = atomicAdd(&cnt[r], 1u);   // pos < CC by construction
          mono[r * CC + pos] = mono_from_float(sc);
          idx[r * CC + pos]  = (unsigned short)gj;
        }
      }
    }
    __syncthreads();                           // inserts done; buffer reusable

    // One ballot picks overflow-risk rows (cnt + 64 > CC). Every wave reads
    // the same 32 counters in the same lane order after the same barrier, so
    // the mask is identical across waves -> uniform sort loop, no deadlock.
    unsigned mask = __builtin_amdgcn_ballot_w32(cnt[r] > (unsigned)(CC - CT));
    while (mask) {
      int rr = __builtin_ctz(mask);
      mask &= mask - 1;
      sort_row_desc(mono, idx, cnt, thr, rr, tid, /*updThr=*/true);
    }
    buf ^= 1;
  }

  // Final per-row ordering + output (scores f32, indices i32).
  for (int rr = 0; rr < ROWS; ++rr)
    sort_row_desc(mono, idx, cnt, thr, rr, tid, /*updThr=*/false);

  for (int e = tid; e < ROWS * KK; e += 256) {
    const int rr = e / KK, s = e - rr * KK;
    const int gi = i0 + rr;
    const unsigned n = cnt[rr];
    float sc; int ix;
    if ((unsigned)s < n) {
      sc = float_from_mono(mono[rr * CC + s]);
      ix = (int)idx[rr * CC + s];
    } else {                            // rows i<50: -inf filled at j=i,i+1,...
      sc = -__builtin_inff();
      ix = gi + (s - (int)n);
    }
    outS[(size_t)gi * KK + s] = sc;
    outI[(size_t)gi * KK + s] = ix;
  }
}

extern "C" void kernel_launch(void* const* d_in, const int* in_sizes, int n_in,
                              void* d_out, int out_size, void* d_ws, size_t ws_size,
                              hipStream_t stream) {
  (void)in_sizes; (void)n_in; (void)d_ws; (void)ws_size; (void)out_size;
  const float* M = (const float*)d_in[0];
  float* outS = (float*)d_out;
  int*   outI = (int*)(outS + (size_t)NN * KK);   // second output region (i32)
  dim3 grid(NN / ROWS), block(256);
  rough_topk_kernel<<<grid, block, SMEM_BYTES, stream>>>(M, outS, outI);
}